// TreeNet_48653389529549
// MI455X (gfx1250) — compile-verified
//
#include <hip/hip_runtime.h>

typedef float v2f __attribute__((ext_vector_type(2)));
typedef float v8f __attribute__((ext_vector_type(8)));

#define NN 100000
#define EE 30000
#define DD 64
#define GQ 2048
#define GW 200
#define EPSF 0.01f

// ---------------- init: out = h (gather node_table by type) ----------------
__global__ void k_init_out(const float* __restrict__ node_table,
                           const int* __restrict__ node_mapping,
                           float* __restrict__ out) {
    int idx = blockIdx.x * blockDim.x + threadIdx.x;
    if (idx >= NN * DD) return;
    int i = idx >> 6, k = idx & 63;
    int ty = node_mapping[2 * i + 1];
    out[idx] = (ty >= 0) ? node_table[(size_t)ty * DD + k] : 0.0f;
}

// ---------------- zero counters ----------------
__global__ void k_zero(int* __restrict__ p, int n) {
    int i = blockIdx.x * blockDim.x + threadIdx.x;
    if (i < n) p[i] = 0;
}

// ---------------- histogram by q and by w ----------------
__global__ void k_hist(const int* __restrict__ rm, int* __restrict__ cnt_q,
                       int* __restrict__ cnt_w) {
    int e = blockIdx.x * blockDim.x + threadIdx.x;
    if (e >= EE) return;
    atomicAdd(&cnt_q[rm[2 * e]], 1);
    atomicAdd(&cnt_w[rm[2 * e + 1]], 1);
}

// ---------------- exclusive scans (edge offsets + tile offsets), one block ----------------
__global__ void k_scan(const int* __restrict__ cnt_q, const int* __restrict__ cnt_w,
                       int* __restrict__ off_q, int* __restrict__ toff_q,
                       int* __restrict__ off_w, int* __restrict__ toff_w) {
    __shared__ int lc[GQ];
    int tid = threadIdx.x;
    for (int i = tid; i < GQ; i += blockDim.x) lc[i] = cnt_q[i];
    __syncthreads();
    if (tid == 0) {
        int eo = 0, to = 0;
        for (int g = 0; g < GQ; g++) {
            off_q[g] = eo; toff_q[g] = to;
            eo += lc[g]; to += (lc[g] + 15) >> 4;
        }
        off_q[GQ] = eo; toff_q[GQ] = to;
    }
    __syncthreads();
    for (int i = tid; i < GW; i += blockDim.x) lc[i] = cnt_w[i];
    __syncthreads();
    if (tid == 0) {
        int eo = 0, to = 0;
        for (int g = 0; g < GW; g++) {
            off_w[g] = eo; toff_w[g] = to;
            eo += lc[g]; to += (lc[g] + 15) >> 4;
        }
        off_w[GW] = eo; toff_w[GW] = to;
    }
}

// ---------------- counting-sort scatter ----------------
__global__ void k_scatter(const int* __restrict__ rm,
                          const int* __restrict__ off_q, int* __restrict__ cur_q,
                          const int* __restrict__ off_w, int* __restrict__ cur_w,
                          int* __restrict__ sorted_q, int* __restrict__ sorted_w) {
    int e = blockIdx.x * blockDim.x + threadIdx.x;
    if (e >= EE) return;
    int q = rm[2 * e], w = rm[2 * e + 1];
    int pq = atomicAdd(&cur_q[q], 1);
    sorted_q[off_q[q] + pq] = e;
    int pw = atomicAdd(&cur_w[w], 1);
    sorted_w[off_w[w] + pw] = e;
}

// ---------------- shared tile helpers ----------------
// Qs padded stride 68 (bank-conflict free for A fragment reads)
// Bs padded stride 17

// pass1: t[:,e] = Q[g] @ h[src_e]  (tiles of 16 edges grouped by q)
__global__ __launch_bounds__(128) void k_pass1(
    const float* __restrict__ Q, const float* __restrict__ h,
    const int* __restrict__ ei, const int* __restrict__ sorted_q,
    const int* __restrict__ off_q, const int* __restrict__ toff_q,
    float* __restrict__ t) {
    int b = blockIdx.x;
    if (b >= toff_q[GQ]) return;
    int lo = 0, hi = GQ;
    while (hi - lo > 1) { int mid = (lo + hi) >> 1; if (toff_q[mid] <= b) lo = mid; else hi = mid; }
    int g = lo;
    int estart = off_q[g] + (b - toff_q[g]) * 16;
    int m = min(16, off_q[g + 1] - estart);

    __shared__ float Qs[64 * 68];
    __shared__ float Bs[64 * 17];
    __shared__ int eids[16];
    __shared__ int srcs[16];
    int tid = threadIdx.x;
    if (tid < 16) {
        int e = (tid < m) ? sorted_q[estart + tid] : -1;
        eids[tid] = e;
        srcs[tid] = (e >= 0) ? ei[e] : -1;
    }
    const float4* Qg = (const float4*)(Q + (size_t)g * 4096);
    for (int i = tid; i < 1024; i += 128) {
        float4 v = Qg[i];
        int r = i >> 4, c4 = (i & 15) * 4;
        float* d = &Qs[r * 68 + c4];
        d[0] = v.x; d[1] = v.y; d[2] = v.z; d[3] = v.w;
    }
    __syncthreads();
    for (int i = tid; i < 256; i += 128) {
        int n = i >> 4, c4 = (i & 15) * 4;
        int s = srcs[n];
        float4 v;
        if (s >= 0) v = *(const float4*)(h + (size_t)s * DD + c4);
        else { v.x = v.y = v.z = v.w = 0.0f; }
        Bs[(c4 + 0) * 17 + n] = v.x; Bs[(c4 + 1) * 17 + n] = v.y;
        Bs[(c4 + 2) * 17 + n] = v.z; Bs[(c4 + 3) * 17 + n] = v.w;
    }
    __syncthreads();

    int lane = tid & 31, wv = tid >> 5;
    int hi16 = lane >> 4, mrow = lane & 15;
    v8f c = {};
    for (int k0 = 0; k0 < 64; k0 += 4) {
        int ka = k0 + (hi16 ? 2 : 0);
        v2f a, bv;
        a.x = Qs[(wv * 16 + mrow) * 68 + ka];
        a.y = Qs[(wv * 16 + mrow) * 68 + ka + 1];
        bv.x = Bs[ka * 17 + mrow];
        bv.y = Bs[(ka + 1) * 17 + mrow];
        c = __builtin_amdgcn_wmma_f32_16x16x4_f32(false, a, false, bv, (short)0, c, false, false);
    }
    if (mrow < m) {
        int e = eids[mrow];
        float* tp = t + (size_t)e * DD + wv * 16 + hi16 * 8;
        #pragma unroll
        for (int v = 0; v < 8; v++) tp[v] = c[v];
    }
}

// pass2: t[:,e] <- t[:,e] + s * X[w>>1] @ t[:,e]  (tiles grouped by w; sign uniform per group)
__global__ __launch_bounds__(128) void k_pass2(
    const float* __restrict__ X, const int* __restrict__ sorted_w,
    const int* __restrict__ off_w, const int* __restrict__ toff_w,
    float* __restrict__ t) {
    int b = blockIdx.x;
    if (b >= toff_w[GW]) return;
    int lo = 0, hi = GW;
    while (hi - lo > 1) { int mid = (lo + hi) >> 1; if (toff_w[mid] <= b) lo = mid; else hi = mid; }
    int g = lo;
    int estart = off_w[g] + (b - toff_w[g]) * 16;
    int m = min(16, off_w[g + 1] - estart);
    float sgn = (g & 1) ? -EPSF : EPSF;

    __shared__ float Qs[64 * 68];
    __shared__ float Bs[64 * 17];
    __shared__ int eids[16];
    int tid = threadIdx.x;
    if (tid < 16) eids[tid] = (tid < m) ? sorted_w[estart + tid] : -1;
    const float4* Xg = (const float4*)(X + (size_t)(g >> 1) * 4096);
    for (int i = tid; i < 1024; i += 128) {
        float4 v = Xg[i];
        int r = i >> 4, c4 = (i & 15) * 4;
        float* d = &Qs[r * 68 + c4];
        d[0] = v.x; d[1] = v.y; d[2] = v.z; d[3] = v.w;
    }
    __syncthreads();
    for (int i = tid; i < 256; i += 128) {
        int n = i >> 4, c4 = (i & 15) * 4;
        int e = eids[n];
        float4 v;
        if (e >= 0) v = *(const float4*)(t + (size_t)e * DD + c4);
        else { v.x = v.y = v.z = v.w = 0.0f; }
        Bs[(c4 + 0) * 17 + n] = v.x; Bs[(c4 + 1) * 17 + n] = v.y;
        Bs[(c4 + 2) * 17 + n] = v.z; Bs[(c4 + 3) * 17 + n] = v.w;
    }
    __syncthreads();

    int lane = tid & 31, wv = tid >> 5;
    int hi16 = lane >> 4, mrow = lane & 15;
    v8f c = {};
    for (int k0 = 0; k0 < 64; k0 += 4) {
        int ka = k0 + (hi16 ? 2 : 0);
        v2f a, bv;
        a.x = Qs[(wv * 16 + mrow) * 68 + ka];
        a.y = Qs[(wv * 16 + mrow) * 68 + ka + 1];
        bv.x = Bs[ka * 17 + mrow];
        bv.y = Bs[(ka + 1) * 17 + mrow];
        c = __builtin_amdgcn_wmma_f32_16x16x4_f32(false, a, false, bv, (short)0, c, false, false);
    }
    if (mrow < m) {
        int e = eids[mrow];
        #pragma unroll
        for (int v = 0; v < 8; v++) {
            int row = wv * 16 + hi16 * 8 + v;
            t[(size_t)e * DD + row] = Bs[row * 17 + mrow] + sgn * c[v];
        }
    }
}

// pass3: msg = Q[g]^T @ u, atomic scatter into out[dst]
__global__ __launch_bounds__(128) void k_pass3(
    const float* __restrict__ Q, const float* __restrict__ t,
    const int* __restrict__ ei, const int* __restrict__ sorted_q,
    const int* __restrict__ off_q, const int* __restrict__ toff_q,
    float* __restrict__ out) {
    int b = blockIdx.x;
    if (b >= toff_q[GQ]) return;
    int lo = 0, hi = GQ;
    while (hi - lo > 1) { int mid = (lo + hi) >> 1; if (toff_q[mid] <= b) lo = mid; else hi = mid; }
    int g = lo;
    int estart = off_q[g] + (b - toff_q[g]) * 16;
    int m = min(16, off_q[g + 1] - estart);

    __shared__ float Qs[64 * 68];
    __shared__ float Bs[64 * 17];
    __shared__ int dsts[16];
    int tid = threadIdx.x;
    if (tid < 16) {
        int e = (tid < m) ? sorted_q[estart + tid] : -1;
        dsts[tid] = (e >= 0) ? ei[EE + e] : -1;
        if (tid < m) {
            // stash edge id via dsts only; B gather uses sorted list directly below
        }
    }
    __shared__ int eids[16];
    if (tid < 16) eids[tid] = (tid < m) ? sorted_q[estart + tid] : -1;
    const float4* Qg = (const float4*)(Q + (size_t)g * 4096);
    for (int i = tid; i < 1024; i += 128) {
        float4 v = Qg[i];
        int r = i >> 4, c4 = (i & 15) * 4;
        float* d = &Qs[r * 68 + c4];
        d[0] = v.x; d[1] = v.y; d[2] = v.z; d[3] = v.w;
    }
    __syncthreads();
    for (int i = tid; i < 256; i += 128) {
        int n = i >> 4, c4 = (i & 15) * 4;
        int e = eids[n];
        float4 v;
        if (e >= 0) v = *(const float4*)(t + (size_t)e * DD + c4);
        else { v.x = v.y = v.z = v.w = 0.0f; }
        Bs[(c4 + 0) * 17 + n] = v.x; Bs[(c4 + 1) * 17 + n] = v.y;
        Bs[(c4 + 2) * 17 + n] = v.z; Bs[(c4 + 3) * 17 + n] = v.w;
    }
    __syncthreads();

    int lane = tid & 31, wv = tid >> 5;
    int hi16 = lane >> 4, mrow = lane & 15;
    v8f c = {};
    for (int k0 = 0; k0 < 64; k0 += 4) {
        int ka = k0 + (hi16 ? 2 : 0);
        v2f a, bv;
        // A = Q^T : A[mrow][k] = Q[k][wv*16+mrow]
        a.x = Qs[ka * 68 + wv * 16 + mrow];
        a.y = Qs[(ka + 1) * 68 + wv * 16 + mrow];
        bv.x = Bs[ka * 17 + mrow];
        bv.y = Bs[(ka + 1) * 17 + mrow];
        c = __builtin_amdgcn_wmma_f32_16x16x4_f32(false, a, false, bv, (short)0, c, false, false);
    }
    if (mrow < m) {
        int d = dsts[mrow];
        float* op = out + (size_t)d * DD + wv * 16 + hi16 * 8;
        #pragma unroll
        for (int v = 0; v < 8; v++) atomicAdd(&op[v], c[v]);
    }
}

extern "C" void kernel_launch(void* const* d_in, const int* in_sizes, int n_in,
                              void* d_out, int out_size, void* d_ws, size_t ws_size,
                              hipStream_t stream) {
    const float* node_table = (const float*)d_in[0];
    const float* X          = (const float*)d_in[1];
    const float* Q          = (const float*)d_in[2];
    const int*   nm         = (const int*)d_in[3];
    const int*   rm         = (const int*)d_in[4];
    const int*   ei         = (const int*)d_in[5];
    float* out = (float*)d_out;

    // workspace layout
    float* t = (float*)d_ws;                       // EE*64 floats
    int* meta = (int*)(t + (size_t)EE * DD);
    int* cnt_q   = meta;                 // 2048
    int* cur_q   = meta + 2048;          // 2048
    int* cnt_w   = meta + 4096;          // 256
    int* cur_w   = meta + 4352;          // 256   (zero first 4608)
    int* off_q   = meta + 4608;          // 2049
    int* toff_q  = meta + 6657;          // 2049
    int* off_w   = meta + 8706;          // 257
    int* toff_w  = meta + 8963;          // 257
    int* sorted_q = meta + 9220;         // EE
    int* sorted_w = sorted_q + EE;       // EE

    k_init_out<<<(NN * DD + 255) / 256, 256, 0, stream>>>(node_table, nm, out);
    k_zero<<<(4608 + 255) / 256, 256, 0, stream>>>(meta, 4608);
    k_hist<<<(EE + 255) / 256, 256, 0, stream>>>(rm, cnt_q, cnt_w);
    k_scan<<<1, 256, 0, stream>>>(cnt_q, cnt_w, off_q, toff_q, off_w, toff_w);
    k_scatter<<<(EE + 255) / 256, 256, 0, stream>>>(rm, off_q, cur_q, off_w, cur_w,
                                                    sorted_q, sorted_w);
    // tile upper bounds: ceil(E/16) + #groups
    const int TQ = (EE + 15) / 16 + GQ;   // 3923
    const int TW = (EE + 15) / 16 + GW;   // 2075
    k_pass1<<<TQ, 128, 0, stream>>>(Q, out, ei, sorted_q, off_q, toff_q, t);
    k_pass2<<<TW, 128, 0, stream>>>(X, sorted_w, off_w, toff_w, t);
    k_pass3<<<TQ, 128, 0, stream>>>(Q, t, ei, sorted_q, off_q, toff_q, out);
}